// AFEModule_7301444403796
// MI455X (gfx1250) — compile-verified
//
#include <hip/hip_runtime.h>
#include <math.h>

// Problem constants
#define BB 8
#define CC 256
#define HH 128
#define WW 128
#define IMG (BB*CC)          // 2048
#define HW  (HH*WW)          // 16384

typedef float v2f __attribute__((ext_vector_type(2)));
typedef float v8f __attribute__((ext_vector_type(8)));

#define V8F_ZERO {0.f,0.f,0.f,0.f,0.f,0.f,0.f,0.f}

// ---- WMMA f32 16x16x4 helpers (wave32) -------------------------------------
// A (16x4): lanes 0-15 -> M=lane, K=kBase+{0,1}; lanes 16-31 -> M=lane-16, K=kBase+{2,3}
__device__ __forceinline__ v2f ld_a(const float* __restrict__ A, int lda,
                                    int mBase, int kBase, int lane) {
    int m = mBase + (lane & 15);
    int k = kBase + ((lane >> 4) << 1);
    const float* p = A + (size_t)m * lda + k;
    v2f r; r.x = p[0]; r.y = p[1];
    return r;
}
// B (4x16) stored row-major as B[k][n]
__device__ __forceinline__ v2f ld_b(const float* __restrict__ Bm, int ldb,
                                    int kBase, int nBase, int lane) {
    int n = nBase + (lane & 15);
    int k = kBase + ((lane >> 4) << 1);
    v2f r;
    r.x = Bm[(size_t)k * ldb + n];
    r.y = Bm[(size_t)(k + 1) * ldb + n];
    return r;
}
// B (4x16) where storage is transposed: B[k][n] = Bt[n][k]
__device__ __forceinline__ v2f ld_bt(const float* __restrict__ Bt, int ldt,
                                     int kBase, int nBase, int lane) {
    int n = nBase + (lane & 15);
    int k = kBase + ((lane >> 4) << 1);
    const float* p = Bt + (size_t)n * ldt + k;
    v2f r; r.x = p[0]; r.y = p[1];
    return r;
}
// D (16x16): VGPR r: lanes 0-15 -> M=mBase+r; lanes 16-31 -> M=mBase+8+r
__device__ __forceinline__ void st_d(float* __restrict__ D, int ldd,
                                     int mBase, int nBase, int lane, v8f c) {
    int n  = nBase + (lane & 15);
    int m0 = mBase + ((lane >> 4) << 3);
#pragma unroll
    for (int r = 0; r < 8; ++r)
        D[(size_t)(m0 + r) * ldd + n] = c[r];
}
__device__ __forceinline__ v8f wmma4(v2f a, v2f b, v8f c) {
    return __builtin_amdgcn_wmma_f32_16x16x4_f32(false, a, false, b,
                                                 (short)0, c, false, false);
}
__device__ __forceinline__ v2f vneg(v2f a) { v2f r; r.x = -a.x; r.y = -a.y; return r; }

// ---- Tensor Data Mover staging (global -> LDS), with portable fallback -----
#if __has_builtin(__builtin_amdgcn_tensor_load_to_lds) && \
    __has_builtin(__builtin_amdgcn_s_wait_tensorcnt)
#define USE_TDM 1
#endif

#ifdef USE_TDM
typedef unsigned int u32x4 __attribute__((ext_vector_type(4)));
typedef int          i32x4 __attribute__((ext_vector_type(4)));
typedef int          i32x8 __attribute__((ext_vector_type(8)));

// One-shot 1-D TDM copy of nfloats contiguous floats into LDS.
// D# per CDNA5 ISA ch.8: group0 = {count=1 | lds_addr | global_addr | type=2},
// group1 = {mask=0, data_size=4B, tensor_dim0=nfloats, tensor_dim1=1,
//           tile_dim0=nfloats, tile_dim1=1, stride0=nfloats}.
__device__ __forceinline__ void tdm_load_1d(const float* gsrc, float* ldsdst,
                                            int nfloats) {
    unsigned long long ga = (unsigned long long)(uintptr_t)gsrc;
    unsigned int lds_off  = (unsigned int)(uintptr_t)ldsdst;  // LDS aperture: low 32b = byte offset
    u32x4 g0;
    g0.x = 1u;                                               // count=1 (user descriptor)
    g0.y = lds_off;                                          // lds_addr
    g0.z = (unsigned int)ga;                                 // global_addr[31:0]
    g0.w = (unsigned int)((ga >> 32) & 0x01FFFFFFull) | (2u << 30); // addr[56:32] | type=2
    i32x8 g1;
    g1[0] = (2 << 16);                                       // data_size = 4 bytes
    g1[1] = (nfloats & 0xFFFF) << 16;                        // tensor_dim0[15:0]
    g1[2] = ((nfloats >> 16) & 0xFFFF) | (1 << 16);          // tensor_dim0[31:16] | tensor_dim1=1
    g1[3] = (nfloats & 0xFFFF) << 16;                        // tensor_dim1[31:16]=0 | tile_dim0
    g1[4] = 1;                                               // tile_dim1=1, tile_dim2=0
    g1[5] = nfloats;                                         // tensor_dim0_stride[31:0]
    g1[6] = 0;
    g1[7] = 0;
    i32x4 z4 = (i32x4){0, 0, 0, 0};
#if __clang_major__ >= 23
    i32x8 z8 = (i32x8){0, 0, 0, 0, 0, 0, 0, 0};
    __builtin_amdgcn_tensor_load_to_lds(g0, g1, z4, z4, z8, 0);
#else
    __builtin_amdgcn_tensor_load_to_lds(g0, g1, z4, z4, 0);
#endif
}
#endif

// Stage nfloats from global to LDS; completes after the following barrier.
__device__ __forceinline__ void stage_to_lds(float* dst, const float* src, int nfloats) {
#ifdef USE_TDM
    if (threadIdx.x == 0) tdm_load_1d(src, dst, nfloats);    // one TDM issue per block
#else
    for (int i = threadIdx.x; i < nfloats; i += blockDim.x) dst[i] = src[i];
#endif
}
__device__ __forceinline__ void stage_wait() {
#ifdef USE_TDM
    __builtin_amdgcn_s_wait_tensorcnt(0);
#endif
    __syncthreads();
}

// ---- K0: DFT matrices (fftshift baked in) ----------------------------------
// mats: [FsRe, FsIm, FiRe, FiIm], each 128x128 row-major.
// Fs[v][w] = exp(-2*pi*i*sigma(v)*w/128)/sqrt(128), sigma(v)=(v+64)&127
// Fi[m][v] = exp(+2*pi*i*m*sigma(v)/128)/sqrt(128)
__global__ void gen_dft_mats(float* __restrict__ mats) {
    int idx = blockIdx.x * blockDim.x + threadIdx.x;
    if (idx >= HW) return;
    int r = idx >> 7, c = idx & 127;
    const float s  = 0.08838834764831845f;          // 1/sqrt(128)
    const float w0 = 6.283185307179586f / 128.f;
    float angS = w0 * (float)(((r + 64) & 127) * c);
    mats[0 * HW + idx] =  cosf(angS) * s;
    mats[1 * HW + idx] = -sinf(angS) * s;
    float angI = w0 * (float)(r * ((c + 64) & 127));
    mats[2 * HW + idx] =  cosf(angI) * s;
    mats[3 * HW + idx] =  sinf(angI) * s;
}

// ---- K1: fused forward DFT2, S = Fs @ (x @ Fs^T), Y kept in LDS ------------
// LDS: Xs(64KB) | Yre(64KB) | Yim(64KB) = 192KB dynamic
__global__ void fwd_dft2(const float* __restrict__ x, const float* __restrict__ mats,
                         float* __restrict__ sre, float* __restrict__ sim) {
    extern __shared__ float smem[];
    float* Xs  = smem;
    float* Yre = smem + HW;
    float* Yim = smem + 2 * HW;
    int img = blockIdx.x;
    const float* FsRe = mats;
    const float* FsIm = mats + HW;
    float* Sre = sre + (size_t)img * HW;
    float* Sim = sim + (size_t)img * HW;
    int lane  = threadIdx.x & 31;
    int mBase = (threadIdx.x >> 5) * 16;

    stage_to_lds(Xs, x + (size_t)img * HW, HW);
    stage_wait();

    // Phase 1: Y = Xs @ Fs^T  (real A from LDS, complex out to LDS)
    for (int nT = 0; nT < 8; ++nT) {
        int nBase = nT * 16;
        v8f aR = V8F_ZERO, aI = V8F_ZERO;
        for (int k = 0; k < 128; k += 4) {
            v2f a   = ld_a (Xs,   128, mBase, k, lane);
            v2f bre = ld_bt(FsRe, 128, k, nBase, lane);  // B[k=w][n=v] = FsRe[v][w]
            v2f bim = ld_bt(FsIm, 128, k, nBase, lane);
            aR = wmma4(a, bre, aR);
            aI = wmma4(a, bim, aI);
        }
        st_d(Yre, 128, mBase, nBase, lane, aR);
        st_d(Yim, 128, mBase, nBase, lane, aI);
    }
    __syncthreads();

    // Phase 2: S = Fs @ Y  (complex x complex, B from LDS, out to global)
    for (int nT = 0; nT < 8; ++nT) {
        int nBase = nT * 16;
        v8f aR = V8F_ZERO, aI = V8F_ZERO;
        for (int k = 0; k < 128; k += 4) {
            v2f are  = ld_a(FsRe, 128, mBase, k, lane);
            v2f aim  = ld_a(FsIm, 128, mBase, k, lane);
            v2f aimn = vneg(aim);
            v2f bre  = ld_b(Yre, 128, k, nBase, lane);
            v2f bim  = ld_b(Yim, 128, k, nBase, lane);
            aR = wmma4(are,  bre, aR);  aR = wmma4(aimn, bim, aR);
            aI = wmma4(aim,  bre, aI);  aI = wmma4(are,  bim, aI);
        }
        st_d(Sre, 128, mBase, nBase, lane, aR);
        st_d(Sim, 128, mBase, nBase, lane, aI);
    }
}

// ---- K2: spectrum-center MLP -> anisotropic 7x7 kernel (one block / batch) -
__global__ void mlp_build_kernel(const float* __restrict__ sre, const float* __restrict__ sim,
                                 const float* __restrict__ w1, const float* __restrict__ b1,
                                 const float* __restrict__ w2, const float* __restrict__ b2,
                                 float* __restrict__ kern) {
    __shared__ float red[256];
    __shared__ float flat[49];
    __shared__ float hmid[32];
    __shared__ float prm[3];
    __shared__ float kv[49];
    __shared__ float ksum;
    int b = blockIdx.x, tid = threadIdx.x;
    for (int p = 0; p < 49; ++p) {                   // mean over C of |S| center 7x7
        int i = p / 7, j = p % 7;
        size_t off = (((size_t)(b * CC + tid)) * 128 + (61 + i)) * 128 + (61 + j);
        float re = sre[off], im = sim[off];
        red[tid] = sqrtf(re * re + im * im);
        __syncthreads();
        for (int s = 128; s > 0; s >>= 1) {
            if (tid < s) red[tid] += red[tid + s];
            __syncthreads();
        }
        if (tid == 0) flat[p] = red[0] * (1.f / 256.f);
        __syncthreads();
    }
    if (tid < 32) {
        float acc = b1[tid];
        for (int p = 0; p < 49; ++p) acc += flat[p] * w1[tid * 49 + p];
        hmid[tid] = fmaxf(acc, 0.f);
    }
    __syncthreads();
    if (tid < 3) {
        float acc = b2[tid];
        for (int k = 0; k < 32; ++k) acc += hmid[k] * w2[tid * 32 + k];
        prm[tid] = acc;
    }
    __syncthreads();
    float theta = atan2f(prm[0], prm[1]) * 0.5f + 1.5707963267948966f;
    float lam1  = expf(prm[2]);
    float lam2  = 1.f / (lam1 + 1e-8f);
    if (tid < 49) {
        float yy = (float)(tid / 7) - 3.f;
        float xx = (float)(tid % 7) - 3.f;
        float ct = cosf(theta), st = sinf(theta);
        float xr =  xx * ct + yy * st;
        float yr = -xx * st + yy * ct;
        kv[tid] = expf(-(xr * xr / (2.f * lam1 * lam1) + yr * yr / (2.f * lam2 * lam2)));
    }
    __syncthreads();
    if (tid == 0) {
        float s = 0.f;
        for (int p = 0; p < 49; ++p) s += kv[p];
        ksum = s + 1e-8f;
    }
    __syncthreads();
    if (tid < 49) kern[b * 49 + tid] = kv[tid] / ksum;
}

// ---- K3: per-batch 7x7 depthwise conv on Re/Im spectra (zero pad) ----------
__global__ void depthwise(const float* __restrict__ sre, const float* __restrict__ sim,
                          const float* __restrict__ kern,
                          float* __restrict__ fre, float* __restrict__ fim) {
    __shared__ float kv[49];
    int blk = blockIdx.x;
    int img = blk >> 6;                       // 64 blocks per image
    int pix = ((blk & 63) << 8) + threadIdx.x;
    int b = img >> 8;
    if (threadIdx.x < 49) kv[threadIdx.x] = kern[b * 49 + threadIdx.x];
    __syncthreads();
    int u = pix >> 7, v = pix & 127;
    const float* Sre = sre + (size_t)img * HW;
    const float* Sim = sim + (size_t)img * HW;
    float ar = 0.f, ai = 0.f;
#pragma unroll
    for (int i = 0; i < 7; ++i) {
        int y = u + i - 3;
        if ((unsigned)y < 128u) {
#pragma unroll
            for (int j = 0; j < 7; ++j) {
                int xsp = v + j - 3;
                if ((unsigned)xsp < 128u) {
                    float kk = kv[i * 7 + j];
                    ar += kk * Sre[y * 128 + xsp];
                    ai += kk * Sim[y * 128 + xsp];
                }
            }
        }
    }
    fre[(size_t)img * HW + pix] = ar;
    fim[(size_t)img * HW + pix] = ai;
}

// ---- K4: channel mix  G[b] = refine_w(256x256) @ F[b](256x16384) -----------
__global__ void chan_mix(const float* __restrict__ Fin, const float* __restrict__ Wmat,
                         float* __restrict__ Gout) {
    int b = blockIdx.y;
    const float* Bm = Fin  + (size_t)b * CC * HW;   // [c][hw], ld = 16384
    float*       D  = Gout + (size_t)b * CC * HW;
    int lane  = threadIdx.x & 31;
    int mBase = (threadIdx.x >> 5) * 32;            // two 16-row strips per wave
    int nBlk  = blockIdx.x * 64;                    // 64 columns per block
    v8f acc[8];
#pragma unroll
    for (int q = 0; q < 8; ++q) acc[q] = (v8f)V8F_ZERO;
    for (int k = 0; k < 256; k += 4) {
        v2f a0 = ld_a(Wmat, 256, mBase,      k, lane);
        v2f a1 = ld_a(Wmat, 256, mBase + 16, k, lane);
        v2f bt[4];
#pragma unroll
        for (int t = 0; t < 4; ++t) bt[t] = ld_b(Bm, HW, k, nBlk + t * 16, lane);
#pragma unroll
        for (int t = 0; t < 4; ++t) {
            acc[t]     = wmma4(a0, bt[t], acc[t]);
            acc[4 + t] = wmma4(a1, bt[t], acc[4 + t]);
        }
    }
#pragma unroll
    for (int t = 0; t < 4; ++t) {
        st_d(D, HW, mBase,      nBlk + t * 16, lane, acc[t]);
        st_d(D, HW, mBase + 16, nBlk + t * 16, lane, acc[4 + t]);
    }
}

// ---- K5: fused inverse DFT2 (real part) + bilinear upsample add ------------
// out = Re( Fi @ (G @ Fi^T) ) + upsample(x_high)
// LDS: Gre|Gim|Zre|Zim = 256KB dynamic
__global__ void inv_dft2(const float* __restrict__ gre, const float* __restrict__ gim,
                         const float* __restrict__ mats,
                         const float* __restrict__ xhigh,
                         float* __restrict__ out) {
    extern __shared__ float smem[];
    float* Gre = smem;
    float* Gim = smem + HW;
    float* Zre = smem + 2 * HW;
    float* Zim = smem + 3 * HW;
    int img = blockIdx.x;
    const float* FiRe = mats + 2 * HW;
    const float* FiIm = mats + 3 * HW;
    const float* XH   = xhigh + (size_t)img * 4096;  // 64x64
    float*       D    = out   + (size_t)img * HW;
    int lane  = threadIdx.x & 31;
    int mBase = (threadIdx.x >> 5) * 16;

    stage_to_lds(Gre, gre + (size_t)img * HW, HW);
    stage_to_lds(Gim, gim + (size_t)img * HW, HW);
    stage_wait();

    // Phase 1: Z = G @ Fi^T  (A from LDS, out to LDS)
    for (int nT = 0; nT < 8; ++nT) {
        int nBase = nT * 16;
        v8f aR = V8F_ZERO, aI = V8F_ZERO;
        for (int k = 0; k < 128; k += 4) {
            v2f are  = ld_a (Gre,  128, mBase, k, lane);
            v2f aim  = ld_a (Gim,  128, mBase, k, lane);
            v2f bre  = ld_bt(FiRe, 128, k, nBase, lane);  // B[k=v][n=m] = Fi[m][v]
            v2f bim  = ld_bt(FiIm, 128, k, nBase, lane);
            v2f bimn = vneg(bim);
            aR = wmma4(are, bre,  aR);  aR = wmma4(aim, bimn, aR);
            aI = wmma4(are, bim,  aI);  aI = wmma4(aim, bre,  aI);
        }
        st_d(Zre, 128, mBase, nBase, lane, aR);
        st_d(Zim, 128, mBase, nBase, lane, aI);
    }
    __syncthreads();

    // Phase 2: out_re = FiRe @ Zre - FiIm @ Zim  (+ upsample add)
    for (int nT = 0; nT < 8; ++nT) {
        int nBase = nT * 16;
        v8f acc = V8F_ZERO;
        for (int k = 0; k < 128; k += 4) {
            v2f are  = ld_a(FiRe, 128, mBase, k, lane);
            v2f aimn = vneg(ld_a(FiIm, 128, mBase, k, lane));
            v2f bre  = ld_b(Zre, 128, k, nBase, lane);
            v2f bim  = ld_b(Zim, 128, k, nBase, lane);
            acc = wmma4(are,  bre, acc);
            acc = wmma4(aimn, bim, acc);
        }
        // half-pixel-center bilinear 64->128 upsample of x_high, fused at store
        int w  = nBase + (lane & 15);
        int m0 = mBase + ((lane >> 4) << 3);
        int jw = w >> 1, x0, x1; float ww0;
        if (w & 1) { x0 = jw;                    x1 = (jw < 63) ? jw + 1 : 63; ww0 = 0.75f; }
        else       { x0 = (jw > 0) ? jw - 1 : 0; x1 = jw;                      ww0 = 0.25f; }
        float ww1 = 1.f - ww0;
#pragma unroll
        for (int r = 0; r < 8; ++r) {
            int h = m0 + r;
            int jh = h >> 1, y0, y1; float wh0;
            if (h & 1) { y0 = jh;                    y1 = (jh < 63) ? jh + 1 : 63; wh0 = 0.75f; }
            else       { y0 = (jh > 0) ? jh - 1 : 0; y1 = jh;                      wh0 = 0.25f; }
            float wh1 = 1.f - wh0;
            float up = wh0 * (ww0 * XH[y0 * 64 + x0] + ww1 * XH[y0 * 64 + x1])
                     + wh1 * (ww0 * XH[y1 * 64 + x0] + ww1 * XH[y1 * 64 + x1]);
            D[h * 128 + w] = acc[r] + up;
        }
    }
}

// ---- host launcher ---------------------------------------------------------
extern "C" void kernel_launch(void* const* d_in, const int* in_sizes, int n_in,
                              void* d_out, int out_size, void* d_ws, size_t ws_size,
                              hipStream_t stream) {
    (void)in_sizes; (void)n_in; (void)out_size; (void)ws_size;
    const float* x_high = (const float*)d_in[0];
    const float* x_low  = (const float*)d_in[1];
    const float* w1 = (const float*)d_in[2];
    const float* b1 = (const float*)d_in[3];
    const float* w2 = (const float*)d_in[4];
    const float* b2 = (const float*)d_in[5];
    const float* rw = (const float*)d_in[6];
    float* outp = (float*)d_out;

    char* ws = (char*)d_ws;
    float* mats = (float*)ws;                         // 4 * 128*128 floats = 256 KB
    float* kern = (float*)(ws + (4 * HW * 4));        // 8*49 floats
    const size_t SZ  = (size_t)IMG * HW * sizeof(float);   // 128 MiB per buffer
    const size_t OFF = (size_t)1 << 20;
    float* bufAre = (float*)(ws + OFF);               // S spectra / mixed G
    float* bufAim = (float*)(ws + OFF + SZ);
    float* bufBre = (float*)(ws + OFF + 2 * SZ);      // depthwise-filtered F
    float* bufBim = (float*)(ws + OFF + 3 * SZ);

    gen_dft_mats    <<<64,  256, 0, stream>>>(mats);
    fwd_dft2        <<<IMG, 256, 3 * HW * sizeof(float), stream>>>(x_low, mats, bufAre, bufAim);
    mlp_build_kernel<<<BB,  256, 0, stream>>>(bufAre, bufAim, w1, b1, w2, b2, kern);
    depthwise       <<<IMG * 64, 256, 0, stream>>>(bufAre, bufAim, kern, bufBre, bufBim);
    chan_mix        <<<dim3(HW / 64, BB), 256, 0, stream>>>(bufBre, rw, bufAre);
    chan_mix        <<<dim3(HW / 64, BB), 256, 0, stream>>>(bufBim, rw, bufAim);
    inv_dft2        <<<IMG, 256, 4 * HW * sizeof(float), stream>>>(bufAre, bufAim, mats, x_high, outp);
}